// SparseEncoderModel_9105330668184
// MI455X (gfx1250) — compile-verified
//
#include <hip/hip_runtime.h>

typedef __attribute__((ext_vector_type(2))) float v2f;
typedef __attribute__((ext_vector_type(8))) float v8f;

#define BATCH     16
#define MODEL_DIM 2048
#define TIME      8192
#define KEEP      64
#define LAT_ELEMS ((size_t)BATCH * KEEP * MODEL_DIM)

// ---------------------------------------------------------------------------
// Phase 1: attn[b][t] = sum_c x[b][c][t] * w[c] + bias
// One wave computes 16 consecutive t via V_WMMA_F32_16X16X4_F32, looping c in
// K=4 steps. A = w broadcast across M rows; B = 4x16 tile of x. All 1 GiB of
// x is read exactly once (coalesced 64B segments per half-wave) -> HBM bound.
// ---------------------------------------------------------------------------
__global__ __launch_bounds__(256) void attn_wmma_kernel(
    const float* __restrict__ x, const float* __restrict__ w,
    const float* __restrict__ bias, float* __restrict__ attn)
{
    const int lane      = threadIdx.x & 31;
    const int waveInBlk = threadIdx.x >> 5;
    const int gw        = blockIdx.x * 8 + waveInBlk;   // global wave id
    const int b         = gw >> 9;                      // 512 t-tiles / batch
    const int t0        = (gw & 511) << 4;

    const float* xb    = x + (size_t)b * MODEL_DIM * TIME;
    const int    n     = lane & 15;
    const int    khalf = (lane >> 4) << 1;   // lanes 0-15 -> K 0,1 ; 16-31 -> K 2,3

    v8f acc = {};
#pragma unroll 8
    for (int c0 = 0; c0 < MODEL_DIM; c0 += 4) {
        // A (16x4 f32): VGPR0 = K0|K2, VGPR1 = K1|K3, value = w[c] for all M rows
        v2f a;
        a.x = w[c0 + khalf];
        a.y = w[c0 + khalf + 1];
        // B (4x16 f32): row K striped across lanes; same half-split as A
        const float* xr = xb + (size_t)(c0 + khalf) * TIME + t0 + n;
        v2f bm;
        bm.x = xr[0];
        bm.y = xr[TIME];
        // 8 args: (neg_a, A, neg_b, B, c_mod, C, reuse_a, reuse_b)
        acc = __builtin_amdgcn_wmma_f32_16x16x4_f32(
            false, a, false, bm, (short)0, acc, false, false);
    }
    // D row M=0: acc[0] of lanes 0..15, N = lane
    if (lane < 16) {
        attn[b * TIME + t0 + n] = acc[0] + bias[0];
    }
}

// ---------------------------------------------------------------------------
// Phase 2: per-batch top-64 of 8192 (sorted desc, smallest-index tie break).
// One block per batch, attn row held in 32 KB of LDS, 64 argmax sweeps.
// ---------------------------------------------------------------------------
__global__ __launch_bounds__(256) void topk_kernel(
    const float* __restrict__ attn, int* __restrict__ ind_out)
{
    __shared__ float sv[TIME];
    __shared__ float rmax[256];
    __shared__ int   ridx[256];

    const int b   = blockIdx.x;
    const int tid = threadIdx.x;

    for (int t = tid; t < TIME; t += 256) sv[t] = attn[b * TIME + t];
    __syncthreads();

    for (int k = 0; k < KEEP; ++k) {
        float best = -__builtin_inff();
        int   bi   = 0x7FFFFFFF;
        for (int t = tid; t < TIME; t += 256) {
            float v = sv[t];
            if (v > best) { best = v; bi = t; }   // t ascending: keeps lowest idx
        }
        rmax[tid] = best; ridx[tid] = bi;
        __syncthreads();
        for (int s = 128; s > 0; s >>= 1) {
            if (tid < s) {
                float ov = rmax[tid + s]; int oi = ridx[tid + s];
                if (ov > rmax[tid] || (ov == rmax[tid] && oi < ridx[tid])) {
                    rmax[tid] = ov; ridx[tid] = oi;
                }
            }
            __syncthreads();
        }
        const int sel = ridx[0];
        if (tid == 0) ind_out[b * KEEP + k] = sel;
        __syncthreads();                  // everyone has read ridx[0]
        if (tid == 0) sv[sel] = -__builtin_inff();
        __syncthreads();
    }
}

// ---------------------------------------------------------------------------
// Phase 3: latents[b][k][c] = x[b][c][idx[b][k]]  (values scale == 1.0 exactly)
// One block per (b,k); coalesced writes, strided reads (~256 MB line traffic).
// ---------------------------------------------------------------------------
__global__ __launch_bounds__(256) void gather_kernel(
    const float* __restrict__ x, const int* __restrict__ ind,
    float* __restrict__ lat)
{
    const int bk = blockIdx.x;            // 0 .. BATCH*KEEP-1
    const int b  = bk / KEEP;
    const int t  = ind[bk];
    const float* xb = x + (size_t)b * MODEL_DIM * TIME + t;
    float* o = lat + (size_t)bk * MODEL_DIM;
    for (int c = threadIdx.x; c < MODEL_DIM; c += 256)
        o[c] = xb[(size_t)c * TIME];
}

extern "C" void kernel_launch(void* const* d_in, const int* in_sizes, int n_in,
                              void* d_out, int out_size, void* d_ws, size_t ws_size,
                              hipStream_t stream)
{
    const float* x    = (const float*)d_in[0];   // [16, 2048, 8192] f32
    const float* w    = (const float*)d_in[1];   // [2048] f32
    const float* bias = (const float*)d_in[2];   // [1] f32

    float* lat     = (float*)d_out;                       // [16, 64, 2048] f32
    int*   ind_out = (int*)((float*)d_out + LAT_ELEMS);   // [16, 64] int32 (tuple tail)
    float* attn    = (float*)d_ws;                        // 16*8192 f32 scratch

    attn_wmma_kernel<<<BATCH * (TIME / 16) / 8, 256, 0, stream>>>(x, w, bias, attn);
    topk_kernel<<<BATCH, 256, 0, stream>>>(attn, ind_out);
    gather_kernel<<<BATCH * KEEP, 256, 0, stream>>>(x, ind_out, lat);
}